// SAGEEncoder_22110491639906
// MI455X (gfx1250) — compile-verified
//
#include <hip/hip_runtime.h>
#include <hip/hip_bf16.h>

typedef __attribute__((ext_vector_type(2))) float v2f;
typedef __attribute__((ext_vector_type(8))) float v8f;

#define DD 128
// packed weight tile in LDS: 64 pair-rows, stride 144 float2 (=288 dwords; 288%64==32
// -> lanes 0-15 (pair p) and 16-31 (pair p+1) hit disjoint 32-bank halves: conflict-free)
#define PR 144
#define WTILE_F32 (64 * PR * 2)   // floats per staged matrix = 18432 (73728 B)

// Stage W[128x128] into packed LDS layout: sW[p*PR + n] = {W[2p][n], W[2p+1][n]}
__device__ __forceinline__ void stage_weights(float* __restrict__ sW,
                                              const float* __restrict__ W)
{
    #pragma unroll 8
    for (int i = 0; i < (DD * DD) / 256; ++i) {
        const int idx = threadIdx.x + i * 256;
        const int k = idx >> 7;
        const int n = idx & 127;
        sW[(k >> 1) * (PR * 2) + 2 * n + (k & 1)] = W[idx];
    }
}

// C = A0 @ W0 [+ A1 @ W1] + bias, optional relu, optional second output copy.
// One wave computes a 16x128 strip of C as 8 16x16 WMMA tiles (f32 WMMA 16x16x4).
// B operands come from LDS (one ds_load_b64 per tile per K-step).
__global__ __launch_bounds__(256) void sage_gemm(
    const float* __restrict__ A0, const float* __restrict__ W0,
    const float* __restrict__ A1, const float* __restrict__ W1,
    const float* __restrict__ bias,
    float* __restrict__ out, float* __restrict__ out2,
    int nrows, int do_relu)
{
    extern __shared__ float smem[];
    float* sW0 = smem;
    float* sW1 = smem + WTILE_F32;

    stage_weights(sW0, W0);
    if (A1) stage_weights(sW1, W1);
    __syncthreads();

    const int lane = threadIdx.x & 31;
    const int wave = threadIdx.x >> 5;
    const int row0 = blockIdx.x * 128 + wave * 16;
    if (row0 < nrows) {                        // wave-uniform; EXEC stays all-1s
        const int mrow   = row0 + (lane & 15); // A: lane L -> row M = L%16
        const int koff   = (lane >> 4) * 2;    // A: VGPR0/1 hold K+koff, K+koff+1
        const int ncol_l = lane & 15;          // B/C: column within 16-wide tile
        const int lh     = lane >> 4;          // B: pair-row offset (K+2*lh, +1)

        v8f acc[8] = {};

        const v2f* bw0 = (const v2f*)sW0;
        const float* aptr = A0 + (size_t)mrow * DD + koff;
        #pragma unroll 4
        for (int k = 0; k < DD; k += 4) {
            v2f a = *(const v2f*)(aptr + k);
            const int pbase = ((k >> 1) + lh) * PR + ncol_l;
            #pragma unroll
            for (int t = 0; t < 8; ++t) {
                v2f b = bw0[pbase + t * 16];
                acc[t] = __builtin_amdgcn_wmma_f32_16x16x4_f32(
                    false, a, false, b, (short)0, acc[t], false, false);
            }
        }
        if (A1) {                              // fused second GEMM (h@Ws + neigh@Wn)
            const v2f* bw1 = (const v2f*)sW1;
            const float* aptr1 = A1 + (size_t)mrow * DD + koff;
            #pragma unroll 4
            for (int k = 0; k < DD; k += 4) {
                v2f a = *(const v2f*)(aptr1 + k);
                const int pbase = ((k >> 1) + lh) * PR + ncol_l;
                #pragma unroll
                for (int t = 0; t < 8; ++t) {
                    v2f b = bw1[pbase + t * 16];
                    acc[t] = __builtin_amdgcn_wmma_f32_16x16x4_f32(
                        false, a, false, b, (short)0, acc[t], false, false);
                }
            }
        }

        // C layout: VGPR r -> rows r (lanes 0-15) and 8+r (lanes 16-31), col = lane%16
        const int srow = row0 + lh * 8;
        #pragma unroll
        for (int t = 0; t < 8; ++t) {
            const int col = t * 16 + ncol_l;
            const float bv = bias[col];
            #pragma unroll
            for (int r = 0; r < 8; ++r) {
                float v = acc[t][r] + bv;
                if (do_relu) v = v > 0.0f ? v : 0.0f;
                const size_t idx = (size_t)(srow + r) * DD + col;
                out[idx] = v;
                if (out2) out2[idx] = v;
            }
        }
    }
}

// segment-max over edges: neigh[dst] = max(neigh[dst], hp[src]) per column.
// hp >= 0 (post-ReLU) so float max == signed-int max on the bit patterns.
__global__ __launch_bounds__(256) void edge_max(
    const float* __restrict__ hp, const int* __restrict__ src,
    const int* __restrict__ dst, int* __restrict__ neigh_i, int nedges)
{
    const int e = blockIdx.x * 8 + (threadIdx.x >> 5);
    if (e >= nedges) return;
    const int lane = threadIdx.x & 31;
    const int s = src[e];
    const int d = dst[e];
    const float4 v = *(const float4*)(hp + (size_t)s * DD + lane * 4);
    int* base = neigh_i + (size_t)d * DD + lane * 4;
    atomicMax(base + 0, __float_as_int(v.x));
    atomicMax(base + 1, __float_as_int(v.y));
    atomicMax(base + 2, __float_as_int(v.z));
    atomicMax(base + 3, __float_as_int(v.w));
}

__global__ void zero_stats(float* __restrict__ stats)
{
    stats[threadIdx.x] = 0.0f;
}

// per-column sum and sum-of-squares over a 256-row chunk, coalesced row-major.
__global__ __launch_bounds__(256) void stats_kernel(
    const float* __restrict__ X, float* __restrict__ stats, int nrows)
{
    __shared__ float ssum[256];
    __shared__ float ssq[256];
    const int col  = threadIdx.x & 127;
    const int rsub = threadIdx.x >> 7;           // 0 or 1: two rows in flight
    const long row0 = (long)blockIdx.x * 256;
    float s = 0.0f, q = 0.0f;
    for (int r = rsub; r < 256; r += 2) {
        const long row = row0 + r;
        if (row < nrows) {
            const float x = X[row * DD + col];
            s += x; q += x * x;
        }
    }
    ssum[threadIdx.x] = s;
    ssq[threadIdx.x]  = q;
    __syncthreads();
    if (threadIdx.x < 128) {
        s = ssum[threadIdx.x] + ssum[threadIdx.x + 128];
        q = ssq[threadIdx.x]  + ssq[threadIdx.x + 128];
        atomicAdd(&stats[col], s);
        atomicAdd(&stats[DD + col], q);
    }
}

__global__ __launch_bounds__(256) void bn_relu_kernel(
    const float* __restrict__ X, const float* __restrict__ stats,
    const float* __restrict__ gamma, const float* __restrict__ beta,
    float* __restrict__ out, long total, float invN)
{
    const long i = (long)blockIdx.x * 256 + threadIdx.x;
    if (i >= total) return;
    const int col = (int)(i & (DD - 1));
    const float mean = stats[col] * invN;
    float var = stats[DD + col] * invN - mean * mean;
    var = var < 0.0f ? 0.0f : var;
    const float sc = gamma[col] * rsqrtf(var + 1e-5f);
    float v = (X[i] - mean) * sc + beta[col];
    out[i] = v > 0.0f ? v : 0.0f;
}

extern "C" void kernel_launch(void* const* d_in, const int* in_sizes, int n_in,
                              void* d_out, int out_size, void* d_ws, size_t ws_size,
                              hipStream_t stream)
{
    const float* feat    = (const float*)d_in[0];
    const float* W_pool  = (const float*)d_in[1];
    const float* b_pool  = (const float*)d_in[2];
    const float* W_self  = (const float*)d_in[3];
    const float* W_neigh = (const float*)d_in[4];
    const float* bias    = (const float*)d_in[5];
    const float* gamma   = (const float*)d_in[6];
    const float* beta    = (const float*)d_in[7];
    const int*   src     = (const int*)d_in[8];
    const int*   dst     = (const int*)d_in[9];

    const int N = in_sizes[0] / DD;       // 100000
    const int E = in_sizes[8];            // 1600000

    const size_t nd = (size_t)N * DD;
    float* h_cur = (float*)d_ws;          // current layer features
    float* hp    = h_cur + nd;            // pool activations; reused as `lin`
    float* neigh = hp + nd;               // segment-max result
    float* stats = neigh + nd;            // 256 floats: [sum | sumsq]
    float* outp  = (float*)d_out;

    const int  gemmGrid  = (N + 127) / 128;
    const int  edgeGrid  = (E + 7) / 8;
    const int  statsGrid = (N + 255) / 256;
    const long total     = (long)nd;
    const int  bnGrid    = (int)((total + 255) / 256);

    const size_t lds1 = (size_t)WTILE_F32 * sizeof(float);      // one staged matrix
    const size_t lds2 = 2 * lds1;                               // two staged matrices

    const float* h = feat;
    for (int l = 0; l < 3; ++l) {
        const float* Wp  = W_pool  + (size_t)l * DD * DD;
        const float* bp  = b_pool  + (size_t)l * DD;
        const float* Wsl = W_self  + (size_t)l * DD * DD;
        const float* Wn  = W_neigh + (size_t)l * DD * DD;
        const float* bl  = bias    + (size_t)l * DD;
        const float* gl  = gamma   + (size_t)l * DD;
        const float* btl = beta    + (size_t)l * DD;

        // hp = relu(h @ Wp + bp); neigh initialized to hp (self-loop)
        sage_gemm<<<gemmGrid, 256, lds1, stream>>>(h, Wp, nullptr, nullptr, bp,
                                                   hp, neigh, N, 1);
        // neigh[d] = max over in-edges of hp[s]
        edge_max<<<edgeGrid, 256, 0, stream>>>(hp, src, dst, (int*)neigh, E);
        // lin = h @ Ws + neigh @ Wn + bias   (lin reuses hp buffer)
        sage_gemm<<<gemmGrid, 256, lds2, stream>>>(h, Wsl, neigh, Wn, bl,
                                                   hp, nullptr, N, 0);
        // batch-norm statistics + fused normalize/affine/relu
        zero_stats<<<1, 256, 0, stream>>>(stats);
        stats_kernel<<<statsGrid, 256, 0, stream>>>(hp, stats, N);
        float* dstp = (l == 2) ? outp : h_cur;
        bn_relu_kernel<<<bnGrid, 256, 0, stream>>>(hp, stats, gl, btl, dstp,
                                                   total, 1.0f / (float)N);
        h = h_cur;
    }
}